// Spiking1DS2NetLayer_8246337208350
// MI455X (gfx1250) — compile-verified
//
#include <hip/hip_runtime.h>
#include <stdint.h>

typedef float v4f __attribute__((ext_vector_type(4)));

// Problem constants (match reference: B=64, C=256, T=2048)
#define B_SZ   64
#define C_CH   256
#define T_LEN  2048

// Tiling
#define ROWS        64                 // channels per block (one "quarter")
#define CHUNK       64                 // timesteps per tile
#define ROW_F4      (CHUNK / 4)        // 16 float4 per row
#define ROW_STRIDE  (CHUNK + 4)        // padded row stride in floats (272B, 16B aligned)
#define NCHUNK      (T_LEN / CHUNK)    // 32
#define LD_PER_THD  16                 // (64 rows * 16 f4) / 64 threads
#define NBUF        3                  // triple-buffered x tiles (2-deep prefetch)

// ---------------------------------------------------------------------------
// CDNA5 async global->LDS copy (ASYNCcnt-tracked). Non-temporal: x is
// streamed exactly once, keep it out of L2.
// ---------------------------------------------------------------------------
__device__ __forceinline__ void async_load16(unsigned lds, unsigned long long gptr) {
    asm volatile("global_load_async_to_lds_b128 %0, %1, off th:TH_LOAD_NT"
                 :: "v"(lds), "v"(gptr)
                 : "memory");
}

// ASYNCcnt here tracks ONLY async loads (writeback uses STOREcnt), and loads
// retire in order -> "<= 16" proves the oldest 16-load batch has landed.
__device__ __forceinline__ void wait_async_le16() {
    asm volatile("s_wait_asynccnt 0x10" ::: "memory");
}
__device__ __forceinline__ void wait_async_0() {
    asm volatile("s_wait_asynccnt 0x0" ::: "memory");
}

// ---------------------------------------------------------------------------
// Kernel 1: per-channel norm precompute.
//   ws[c]     = b[c] * norm[c]          (reset magnitude)
//   ws[C + c] = 1 / (norm[c] + 1e-8)    (threshold scale)
// conv_weight is (C, C, 3) row-major -> 768 contiguous floats per out-channel.
// ---------------------------------------------------------------------------
__global__ void lif_norm_kernel(const float* __restrict__ w,
                                const float* __restrict__ b,
                                float* __restrict__ ws) {
    const int c = threadIdx.x;                       // 256 threads, 1 block
    const v4f* wr = (const v4f*)(w + (size_t)c * (C_CH * 3));
    float s = 0.0f;
#pragma unroll 4
    for (int i = 0; i < (C_CH * 3) / 4; ++i) {       // 192 b128 loads
        v4f v = wr[i];
        s += v.x * v.x + v.y * v.y + v.z * v.z + v.w * v.w;
    }
    ws[c]        = b[c] * s;
    ws[C_CH + c] = 1.0f / (s + 1e-8f);
}

// ---------------------------------------------------------------------------
// Kernel 2: LIF scan. Grid = 256 blocks (64 batches x 4 channel-quarters),
// 64 threads/block (2 waves, wave32). Thread owns channel c = q*64 + tid.
// Triple-buffered async LDS staging of x tiles; coalesced NT b128 writeback.
// ---------------------------------------------------------------------------
__global__ void __launch_bounds__(64)
lif_scan_kernel(const float* __restrict__ x,
                const float* __restrict__ bvec,
                const float* __restrict__ betap,
                const float* __restrict__ ws,
                float* __restrict__ out) {
    __shared__ float xbuf[NBUF][ROWS * ROW_STRIDE];  // 3 x 17KB
    __shared__ float obuf[ROWS * ROW_STRIDE];        // 17KB

    const int tid = threadIdx.x;                     // 0..63
    const int bi  = blockIdx.x >> 2;                 // batch
    const int q   = blockIdx.x & 3;                  // channel quarter
    const int c0  = q * ROWS;
    const int c   = c0 + tid;

    const float beta = betap[0];
    const float omb  = 1.0f - beta;
    const float bn   = ws[c];                        // b*norm
    const float invn = ws[C_CH + c];                 // 1/(norm+eps)
    const float bb   = bvec[c];

    // Base flat index of this block's tile: row r -> x[(bi*C + c0 + r)*T + t]
    const size_t gbase = ((size_t)bi * C_CH + c0) * (size_t)T_LEN;

    // Transposed cooperative mapping: unit j -> row = 4j + (tid>>4), col = tid&15.
    // A wave covers two full 256B-contiguous global rows per unit -> coalesced.
    const int mrow = tid >> 4;                       // 0..3
    const int mcol = tid & 15;                       // float4 column

    float mem = 0.0f, spk = 0.0f;

    // Prefetch chunks 0 and 1 (NCHUNK >= 2).
#pragma unroll
    for (int p = 0; p < 2; ++p) {
        const float* gp = x + gbase + (size_t)(p * CHUNK);
#pragma unroll
        for (int j = 0; j < LD_PER_THD; ++j) {
            const int row = 4 * j + mrow;
            const unsigned lds =
                (unsigned)(size_t)&xbuf[p][row * ROW_STRIDE + mcol * 4];
            async_load16(lds, (unsigned long long)(size_t)(gp + (size_t)row * T_LEN + mcol * 4));
        }
    }

    int cur = 0;                                     // buffer holding chunk k
    for (int k = 0; k < NCHUNK; ++k) {
        const int t0 = k * CHUNK;

        // In flight here: chunk k (oldest, 16 loads) and maybe chunk k+1.
        if (k + 1 < NCHUNK) wait_async_le16();       // chunk k landed, k+1 in flight
        else                wait_async_0();
        __syncthreads();          // both waves' loads visible; obuf reusable

        // Issue chunk k+2 into the buffer last used by chunk k-1 (free now).
        if (k + 2 < NCHUNK) {
            const int nb = (cur + 2 >= NBUF) ? (cur + 2 - NBUF) : (cur + 2);
            const float* gp = x + gbase + (size_t)(t0 + 2 * CHUNK);
#pragma unroll
            for (int j = 0; j < LD_PER_THD; ++j) {
                const int row = 4 * j + mrow;
                const unsigned lds =
                    (unsigned)(size_t)&xbuf[nb][row * ROW_STRIDE + mcol * 4];
                async_load16(lds, (unsigned long long)(size_t)(gp + (size_t)row * T_LEN + mcol * 4));
            }
        }

        // Serial LIF recurrence over this chunk, own channel row (row = tid).
        {
            const v4f* xr   = (const v4f*)&xbuf[cur][tid * ROW_STRIDE];
            v4f*       orow = (v4f*)&obuf[tid * ROW_STRIDE];
#pragma unroll
            for (int j = 0; j < ROW_F4; ++j) {
                const v4f xv = xr[j];                // ds_load_b128
                v4f sv;
                { mem = (mem - spk * bn) * beta + xv.x * omb;
                  spk = (mem * invn - bb) > 0.0f ? 1.0f : 0.0f; sv.x = spk; }
                { mem = (mem - spk * bn) * beta + xv.y * omb;
                  spk = (mem * invn - bb) > 0.0f ? 1.0f : 0.0f; sv.y = spk; }
                { mem = (mem - spk * bn) * beta + xv.z * omb;
                  spk = (mem * invn - bb) > 0.0f ? 1.0f : 0.0f; sv.z = spk; }
                { mem = (mem - spk * bn) * beta + xv.w * omb;
                  spk = (mem * invn - bb) > 0.0f ? 1.0f : 0.0f; sv.w = spk; }
                orow[j] = sv;                        // ds_store_b128
            }
        }

        __syncthreads();          // out-tile complete across both waves

        // Coalesced non-temporal writeback (same transposed mapping).
        {
            float* gw = out + gbase + (size_t)t0;
#pragma unroll
            for (int j = 0; j < LD_PER_THD; ++j) {
                const int row = 4 * j + mrow;
                const v4f v = *(const v4f*)&obuf[row * ROW_STRIDE + mcol * 4];
                __builtin_nontemporal_store(v, (v4f*)(gw + (size_t)row * T_LEN + mcol * 4));
            }
        }

        cur = (cur + 1 >= NBUF) ? 0 : (cur + 1);
    }
}

// ---------------------------------------------------------------------------
// Host launcher. Inputs (reference dict order): x, conv_weight, beta, b.
// d_ws: needs 512 floats (2KB) for {b*norm, 1/(norm+eps)}.
// ---------------------------------------------------------------------------
extern "C" void kernel_launch(void* const* d_in, const int* in_sizes, int n_in,
                              void* d_out, int out_size, void* d_ws, size_t ws_size,
                              hipStream_t stream) {
    const float* x    = (const float*)d_in[0];
    const float* w    = (const float*)d_in[1];
    const float* beta = (const float*)d_in[2];
    const float* b    = (const float*)d_in[3];
    float* out = (float*)d_out;
    float* ws  = (float*)d_ws;

    lif_norm_kernel<<<dim3(1), dim3(C_CH), 0, stream>>>(w, b, ws);
    lif_scan_kernel<<<dim3(B_SZ * 4), dim3(ROWS), 0, stream>>>(x, b, beta, ws, out);
}